// TotalLoss_vis_atten_thm_4_23991687315989
// MI455X (gfx1250) — compile-verified
//
#include <hip/hip_runtime.h>
#include <hip/hip_bf16.h>
#include <math.h>

#define OUT_H 152
#define OUT_W 272
#define HW    (OUT_H * OUT_W)        // 41344
#define NBLK  ((HW + 255) / 256)     // 162 focal partial blocks per batch
#define NANN  64
#define N_ID  537
#define EMB   256
#define BATCH 8
#define NPAD  544                    // 34 * 16 padded class dim

// ---- workspace layout (float offsets into d_ws) ----
#define F_OBJP 0        // 8*64*4  : ctxf, ctyf, radius, inv2sig2
#define F_IND  2048     // 8*64    : int gather indices
#define F_MASK 2560     // 8*64
#define F_V    3072     // 8*64
#define F_IDS  3584     // 8*64    : int target ids
#define F_WH   4096     // 8*64*8
#define F_PART 8192     // 8*162*3 : focal partials (pos, neg, npos)
#define F_LOSS 12288    // 4*8     : cls/reg/vis/id per batch
#define F_FEAT 12544    // 8*64*256 normalized features
#define F_LOG  143616   // 8*64*544 logits
#define F_ZERO 422144   // 256-float zero row (B operand for padded cols)
// total 422400 floats = 1.69 MB

typedef __attribute__((ext_vector_type(2))) float v2f;
typedef __attribute__((ext_vector_type(8))) float v8f;

__device__ __forceinline__ float bred256(float v, float* s) {
  int tid = threadIdx.x;
  __syncthreads();
  s[tid] = v;
  __syncthreads();
  for (int k = 128; k > 0; k >>= 1) {
    if (tid < k) s[tid] = s[tid] + s[tid + k];
    __syncthreads();
  }
  return s[0];
}

// ---------------- kernel 1: build per-object targets ----------------
__global__ void k_targets(const float* __restrict__ ann1,
                          const float* __restrict__ ann2,
                          float* __restrict__ ws) {
  int b = blockIdx.x;
  int o = threadIdx.x;            // 64 threads
  int* wi = (int*)ws;

  // block 0 also clears the zero row used by the GEMM's padded columns
  if (b == 0) {
    for (int i = 0; i < 4; ++i) ws[F_ZERO + o * 4 + i] = 0.f;
  }

  const float* a1 = ann1 + (b * NANN + o) * 8;
  const float* A2 = ann2 + b * NANN * 8;

  float cx_raw = a1[0] * (float)OUT_W;
  float cy_raw = a1[1] * (float)OUT_H;
  float w = a1[2] * (float)OUT_W;
  float h = a1[3] * (float)OUT_H;
  float x1 = cx_raw - 0.5f * w, y1 = cy_raw - 0.5f * h;
  float x2 = x1 + w, y2 = y1 + h;
  float cx = fminf(fmaxf(cx_raw, 0.f), (float)OUT_W - 1.f);
  float cy = fminf(fmaxf(cy_raw, 0.f), (float)OUT_H - 1.f);
  int ctx = (int)cx, cty = (int)cy;

  // gaussian_radius(ceil(h), ceil(w)) with min_overlap = 0.7
  float hh = ceilf(h), wc = ceilf(w);
  float b1 = hh + wc;
  float c1 = wc * hh * (1.f - 0.7f) / (1.f + 0.7f);
  float r1 = (b1 + sqrtf(fmaxf(b1 * b1 - 4.f * c1, 0.f))) * 0.5f;
  float b2 = 2.f * (hh + wc);
  float c2 = (1.f - 0.7f) * wc * hh;
  float r2 = (b2 + sqrtf(fmaxf(b2 * b2 - 16.f * c2, 0.f))) * 0.5f;
  float b3 = -2.f * 0.7f * (hh + wc);
  float c3 = (0.7f - 1.f) * wc * hh;
  float r3 = (b3 + sqrtf(fmaxf(b3 * b3 - 16.f * 0.7f * c3, 0.f))) * 0.5f;
  float radius = fmaxf(floorf(fminf(fminf(r1, r2), r3)), 0.f);
  float sigma = (2.f * radius + 1.f) / 6.f;
  float inv2s2 = 1.f / (2.f * sigma * sigma);

  // first pid match in ann2 (== jnp.argmax over boolean match)
  float pid = a1[5];
  int j = -1;
  for (int k = 0; k < NANN; ++k) {
    if (A2[k * 8 + 5] == pid) { j = k; break; }
  }
  bool has = (j >= 0);
  const float* t1 = A2 + (has ? j : 0) * 8;
  float cx2 = t1[0] * (float)OUT_W, cy2 = t1[1] * (float)OUT_H;
  float w2 = t1[2] * (float)OUT_W,  h2 = t1[3] * (float)OUT_H;
  float x1t = cx2 - 0.5f * w2, y1t = cy2 - 0.5f * h2;
  float x2t = x1t + w2, y2t = y1t + h2;

  bool valid = has && (h > 0.f) && (w > 0.f) && (h2 > 0.f) && (w2 > 0.f);
  float vm = valid ? 1.f : 0.f;
  float cxf = (float)ctx, cyf = (float)cty;

  float* objp = ws + F_OBJP + (b * NANN + o) * 4;
  objp[0] = cxf; objp[1] = cyf; objp[2] = radius; objp[3] = inv2s2;

  wi[F_IND + b * NANN + o] = valid ? (cty * OUT_W + ctx) : 0;
  ws[F_MASK + b * NANN + o] = vm;
  ws[F_V + b * NANN + o] = a1[6] * vm;
  wi[F_IDS + b * NANN + o] = valid ? (int)pid : 0;

  float* wh = ws + F_WH + (b * NANN + o) * 8;
  wh[0] = (cxf - x1) * vm;  wh[1] = (x2 - cxf) * vm;
  wh[2] = (cyf - y1) * vm;  wh[3] = (y2 - cyf) * vm;
  wh[4] = (cxf - x1t) * vm; wh[5] = (x2t - cxf) * vm;
  wh[6] = (cyf - y1t) * vm; wh[7] = (y2t - cyf) * vm;
}

// ------- kernel 2: fused heatmap construction + focal loss partials -------
__global__ void k_focal(const float* __restrict__ cls,
                        float* __restrict__ ws) {
  __shared__ float sobj[NANN * 4];
  __shared__ float sred[256];
  int blk = blockIdx.x, b = blockIdx.y, tid = threadIdx.x;
  sobj[tid] = ws[F_OBJP + b * NANN * 4 + tid];   // 256 = 64*4 params
  __syncthreads();

  int p = blk * 256 + tid;
  float pl = 0.f, nl = 0.f, np = 0.f;
  if (p < HW) {
    float fx = (float)(p % OUT_W);
    float fy = (float)(p / OUT_W);
    float hm = 0.f;
    #pragma unroll 4
    for (int o = 0; o < NANN; ++o) {
      float dx = fx - sobj[o * 4 + 0];
      float dy = fy - sobj[o * 4 + 1];
      float rad = sobj[o * 4 + 2];
      if (fabsf(dy) <= rad && fabsf(dx) <= rad) {
        float g = __expf(-(dy * dy + dx * dx) * sobj[o * 4 + 3]);
        hm = fmaxf(hm, g);
      }
    }
    float xv = cls[b * HW + p];
    float pred = 1.f / (1.f + __expf(-xv));
    pred = fminf(fmaxf(pred, 1e-4f), 1.f - 1e-4f);
    if (hm == 1.f) {
      float om = 1.f - pred;
      pl = __logf(pred) * om * om;
      np = 1.f;
    } else {
      float t = 1.f - hm;
      float t4 = (t * t) * (t * t);
      nl = __logf(1.f - pred) * pred * pred * t4;
    }
  }
  pl = bred256(pl, sred);
  nl = bred256(nl, sred);
  np = bred256(np, sred);
  if (tid == 0) {
    float* part = ws + F_PART + (b * NBLK + blk) * 3;
    part[0] = pl; part[1] = nl; part[2] = np;
  }
}

// ------- kernel 3: gather ID features at ind + L2 normalize (wave/object) ----
__global__ void k_gather(const float* __restrict__ identis,
                         float* __restrict__ ws) {
  int b = blockIdx.x >> 3;
  int grp = blockIdx.x & 7;
  int wave = threadIdx.x >> 5, lane = threadIdx.x & 31;
  int o = grp * 8 + wave;                       // 0..63
  const int* wi = (const int*)ws;
  int idx = wi[F_IND + b * NANN + o];
  float vals[8];
  float ss = 0.f;
  #pragma unroll
  for (int i = 0; i < 8; ++i) {
    int d = i * 32 + lane;
    float x = identis[((size_t)b * EMB + d) * HW + idx];
    vals[i] = x;
    ss += x * x;
  }
  #pragma unroll
  for (int s = 16; s > 0; s >>= 1) ss += __shfl_xor(ss, s, 32);
  float scale = (sqrtf(2.f) * __logf((float)N_ID - 1.f)) /
                fmaxf(sqrtf(ss), 1e-12f);
  float* frow = ws + F_FEAT + ((size_t)b * NANN + o) * EMB;
  #pragma unroll
  for (int i = 0; i < 8; ++i) frow[i * 32 + lane] = vals[i] * scale;
}

// ------- kernel 4: logits = feat @ W_id^T via V_WMMA_F32_16X16X4_F32 -------
__global__ void k_gemm(const float* __restrict__ Wid,
                       float* __restrict__ ws) {
  int b = blockIdx.x;
  int wave = threadIdx.x >> 5;
  int lane = threadIdx.x & 31;
  int lane15 = lane & 15;
  int laneHalf = lane >> 4;

  const float* featB = ws + F_FEAT + (size_t)b * NANN * EMB;
  float* logB = ws + F_LOG + (size_t)b * NANN * NPAD;
  const float* zrow = ws + F_ZERO;

  const int MT = NANN / 16;        // 4
  const int NT = NPAD / 16;        // 34
  for (int tile = wave; tile < MT * NT; tile += 8) {
    int mt = tile / NT, nt = tile % NT;
    int row = mt * 16 + lane15;                  // A: M = lane (per half)
    int col = nt * 16 + lane15;                  // B: N = lane (per half)
    // A fragment: feat[row][k0 + 2*laneHalf + {0,1}]
    const float* arow = featB + row * EMB + 2 * laneHalf;
    // B fragment: W[col][k0 + 2*laneHalf + {0,1}]; padded cols read a zero row
    const float* brow =
        ((col < N_ID) ? (Wid + (size_t)col * EMB) : zrow) + 2 * laneHalf;

    v8f acc = {};
    #pragma unroll 8
    for (int k0 = 0; k0 < EMB; k0 += 4) {
      v2f a  = *(const v2f*)(arow + k0);
      v2f bb = *(const v2f*)(brow + k0);
      acc = __builtin_amdgcn_wmma_f32_16x16x4_f32(
          false, a, false, bb, (short)0, acc, false, false);
    }
    // C/D layout: VGPR v -> M = v (lanes 0-15) or v+8 (lanes 16-31), N = lane
    #pragma unroll
    for (int v = 0; v < 8; ++v) {
      int r = mt * 16 + v + 8 * laneHalf;
      logB[r * NPAD + col] = acc[v];
    }
  }
}

// ------- kernel 5: finalize focal + reg smooth-L1 + vis losses -------
__global__ void k_small(const float* __restrict__ regs,
                        const float* __restrict__ viss,
                        float* __restrict__ ws) {
  __shared__ float sred[256];
  int b = blockIdx.x, tid = threadIdx.x;
  const int* wi = (const int*)ws;

  // focal partial accumulation (fixed order)
  const float* part = ws + F_PART + b * NBLK * 3;
  float pl = 0.f, nl = 0.f, np = 0.f;
  for (int i = tid; i < NBLK; i += 256) {
    pl += part[i * 3 + 0]; nl += part[i * 3 + 1]; np += part[i * 3 + 2];
  }
  pl = bred256(pl, sred);
  nl = bred256(nl, sred);
  np = bred256(np, sred);

  // reg smooth-L1 over 64 objects x 8 channels
  float rsum = 0.f;
  for (int idx = tid; idx < NANN * 8; idx += 256) {
    int o = idx >> 3, ch = idx & 7;
    float m = ws[F_MASK + b * NANN + o];
    int gi = wi[F_IND + b * NANN + o];
    float pred = regs[((size_t)b * 8 + ch) * HW + gi];
    float d = fabsf(pred * m - ws[F_WH + (b * NANN + o) * 8 + ch] * m);
    rsum += (d < 1.f) ? 0.5f * d * d : d - 0.5f;
  }
  rsum = bred256(rsum, sred);

  // vis loss (1 channel, target v)
  float vsum = 0.f, msum = 0.f;
  if (tid < NANN) {
    int o = tid;
    float m = ws[F_MASK + b * NANN + o];
    int gi = wi[F_IND + b * NANN + o];
    float pv = 1.f / (1.f + __expf(-viss[(size_t)b * HW + gi]));
    pv = fminf(fmaxf(pv, 1e-4f), 1.f - 1e-4f);
    float d = fabsf(pv * m - ws[F_V + b * NANN + o] * m);
    vsum = (d < 1.f) ? 0.5f * d * d : d - 0.5f;
    msum = m;
  }
  vsum = bred256(vsum, sred);
  msum = bred256(msum, sred);

  if (tid == 0) {
    float cls_l = (np > 0.f) ? -(pl + nl) / fmaxf(np, 1.f) : -nl;
    ws[F_LOSS + 0 * 8 + b] = cls_l;
    ws[F_LOSS + 1 * 8 + b] = rsum / (msum + 1e-4f);
    ws[F_LOSS + 2 * 8 + b] = vsum / (msum + 1e-4f);
  }
}

// ------- kernel 6: log-softmax + NLL id loss (wave per object) -------
__global__ void k_idloss(const float* __restrict__ b_id,
                         float* __restrict__ ws) {
  __shared__ float swave[8];
  int b = blockIdx.x;
  int wave = threadIdx.x >> 5, lane = threadIdx.x & 31;
  const int* wi = (const int*)ws;
  const float* logB = ws + F_LOG + (size_t)b * NANN * NPAD;

  float wacc = 0.f;
  for (int o = wave; o < NANN; o += 8) {
    const float* lrow = logB + o * NPAD;
    float m = -1e30f;
    for (int c = lane; c < N_ID; c += 32) m = fmaxf(m, lrow[c] + b_id[c]);
    #pragma unroll
    for (int s = 16; s > 0; s >>= 1) m = fmaxf(m, __shfl_xor(m, s, 32));
    float se = 0.f;
    for (int c = lane; c < N_ID; c += 32) se += __expf(lrow[c] + b_id[c] - m);
    #pragma unroll
    for (int s = 16; s > 0; s >>= 1) se += __shfl_xor(se, s, 32);
    float lse = m + __logf(se);
    int tgt = wi[F_IDS + b * NANN + o];
    float lt = lrow[tgt] + b_id[tgt];
    float nll = -ws[F_V + b * NANN + o] * (lt - lse);
    wacc += nll * ws[F_MASK + b * NANN + o];
  }
  if (lane == 0) swave[wave] = wacc;
  __syncthreads();
  if (threadIdx.x == 0) {
    float tot = 0.f, msum = 0.f;
    for (int w = 0; w < 8; ++w) tot += swave[w];
    for (int o = 0; o < NANN; ++o) msum += ws[F_MASK + b * NANN + o];
    ws[F_LOSS + 3 * 8 + b] = tot / fmaxf(msum, 1.f);
  }
}

// ------- kernel 7: batch means + uncertainty passthrough -------
__global__ void k_final(const float* __restrict__ ws,
                        const float* __restrict__ s_det,
                        const float* __restrict__ s_id,
                        float* __restrict__ out) {
  for (int j = 0; j < 4; ++j) {
    float s = 0.f;
    for (int b = 0; b < BATCH; ++b) s += ws[F_LOSS + j * 8 + b];
    out[j] = s / (float)BATCH;
  }
  out[4] = s_det[0];
  out[5] = s_id[0];
}

extern "C" void kernel_launch(void* const* d_in, const int* in_sizes, int n_in,
                              void* d_out, int out_size, void* d_ws, size_t ws_size,
                              hipStream_t stream) {
  (void)in_sizes; (void)n_in; (void)out_size; (void)ws_size;
  const float* cls     = (const float*)d_in[0];
  const float* regs    = (const float*)d_in[1];
  const float* viss    = (const float*)d_in[2];
  const float* identis = (const float*)d_in[3];
  const float* ann1    = (const float*)d_in[4];
  const float* ann2    = (const float*)d_in[5];
  const float* Wid     = (const float*)d_in[6];
  const float* b_id    = (const float*)d_in[7];
  const float* s_det   = (const float*)d_in[8];
  const float* s_id    = (const float*)d_in[9];
  float* out = (float*)d_out;
  float* ws  = (float*)d_ws;

  k_targets<<<BATCH, NANN, 0, stream>>>(ann1, ann2, ws);
  k_focal<<<dim3(NBLK, BATCH), 256, 0, stream>>>(cls, ws);
  k_gather<<<BATCH * 8, 256, 0, stream>>>(identis, ws);
  k_gemm<<<BATCH, 256, 0, stream>>>(Wid, ws);
  k_small<<<BATCH, 256, 0, stream>>>(regs, viss, ws);
  k_idloss<<<BATCH, 256, 0, stream>>>(b_id, ws);
  k_final<<<1, 1, 0, stream>>>(ws, s_det, s_id, out);
}